// Voxel2Point_48584670053112
// MI455X (gfx1250) — compile-verified
//
#include <hip/hip_runtime.h>

// CDNA5 / gfx1250 three-NN + three-interpolate.
// Distance tiles computed with V_WMMA_F32_16X16X4_F32:
//   score(q,t) = |q|^2 - 2 q.t   (A = -2*q rows, B = target cols, C = |q|^2)
// which ranks identically to squared distance for a fixed target.

typedef __attribute__((ext_vector_type(2))) float v2f;
typedef __attribute__((ext_vector_type(8))) float v8f;
typedef __attribute__((ext_vector_type(4))) int   v4i;

#define M_Q     16384          // queries (voxel centers)
#define N_T     131072         // target points
#define CHANS   64
#define CHUNK_Q 2048           // queries staged in LDS per pass
#define NCHUNK  (M_Q / CHUNK_Q)
#define NTILE   (CHUNK_Q / 16)

#if defined(__AMDGCN__) && __has_builtin(__builtin_amdgcn_wmma_f32_16x16x4_f32)
#define USE_WMMA 1
#else
#define USE_WMMA 0
#endif

#if defined(__AMDGCN__) && \
    __has_builtin(__builtin_amdgcn_global_load_async_to_lds_b128) && \
    __has_builtin(__builtin_amdgcn_s_wait_asynccnt)
#define USE_ASYNC 1
// builtin signature: (v4i addrspace(1)* gsrc, v4i addrspace(3)* ldst, imm off, imm cpol)
#define GPTR(p) ((__attribute__((address_space(1))) v4i*)(p))
#define LPTR(p) ((__attribute__((address_space(3))) v4i*)(p))
#else
#define USE_ASYNC 0
#endif

// Sorted insert of (c,ci) into ascending triple (d0,i0)<=(d1,i1)<=(d2,i2).
__device__ __forceinline__ void insert3(float c, int ci,
                                        float &d0, float &d1, float &d2,
                                        int &i0, int &i1, int &i2) {
  bool a1 = c >= d1;
  float h1 = a1 ? c : d1;  int h1i = a1 ? ci : i1;
  bool a0 = c >= d0;
  float h0 = a0 ? c : d0;  int h0i = a0 ? ci : i0;
  bool k2 = h1 < d2;  d2 = k2 ? h1 : d2;  i2 = k2 ? h1i : i2;
  bool k1 = h0 < d1;  d1 = k1 ? h0 : d1;  i1 = k1 ? h0i : i1;
  bool k0 = c  < d0;  d0 = k0 ? c  : d0;  i0 = k0 ? ci  : i0;
}

__device__ __forceinline__ float min8(v8f D) {
  return fminf(fminf(fminf(D[0], D[1]), fminf(D[2], D[3])),
               fminf(fminf(D[4], D[5]), fminf(D[6], D[7])));
}

// indices (M,4) int -> Qs[m] = -2 * (b, x,y,z metric), Qn[m] = |q|^2
__global__ void prep_queries_kernel(const int* __restrict__ ind,
                                    float* __restrict__ Qs,
                                    float* __restrict__ Qn) {
  int m = blockIdx.x * 256 + threadIdx.x;
  if (m >= M_Q) return;
  const float unit = 3.2f / 64.0f;
  const float off  = 0.5f * unit - 1.6f;
  float b = (float)ind[m * 4 + 0];
  float x = (float)ind[m * 4 + 1] * unit + off;
  float y = (float)ind[m * 4 + 2] * unit + off;
  float z = (float)ind[m * 4 + 3] * unit + off;
  float4 q;
  q.x = -2.0f * b; q.y = -2.0f * x; q.z = -2.0f * y; q.w = -2.0f * z;
  ((float4*)Qs)[m] = q;
  Qn[m] = b * b + x * x + y * y + z * z;
}

__global__ __launch_bounds__(256) void three_nn_interp_kernel(
    const float* __restrict__ feats,   // (M_Q, 64)
    const float* __restrict__ pc,      // (N_T, 3)
    const int*   __restrict__ bids,    // (N_T)
    const float* __restrict__ Qs,      // (M_Q, 4)  -2*coords
    const float* __restrict__ Qn,      // (M_Q)     |q|^2
    float* __restrict__ out) {         // (N_T, 64)
  __shared__ float sQs[CHUNK_Q * 4];   // 32 KB
  __shared__ float sQn[CHUNK_Q];       //  8 KB

  const int tid  = threadIdx.x;
  const int wave = tid >> 5;           // 8 waves/block
  const int lane = tid & 31;
  const int lm   = lane & 15;
  const int half = lane >> 4;

  // Each wave owns 16 targets; lane j and j+16 share target column j.
  const int n = blockIdx.x * 128 + wave * 16 + lm;

  float tb = (float)bids[n];
  float tx = pc[n * 3 + 0];
  float ty = pc[n * 3 + 1];
  float tz = pc[n * 3 + 2];
  float tnorm = tb * tb + tx * tx + ty * ty + tz * tz;

  // B operand (4x16 f32): col = lane%16, K = 2*(lane>=16) + vgpr
  v2f Bop;
  Bop.x = half ? ty : tb;
  Bop.y = half ? tz : tx;

  float d0 = 3.0e38f, d1 = 3.0e38f, d2 = 3.0e38f;
  int   i0 = 0, i1 = 0, i2 = 0;

  for (int c = 0; c < NCHUNK; ++c) {
    __syncthreads();
#if USE_ASYNC
    {
      const float4* g4 = (const float4*)Qs + c * CHUNK_Q;
      float4* s4 = (float4*)sQs;
#pragma unroll
      for (int i = 0; i < CHUNK_Q / 256; ++i)
        __builtin_amdgcn_global_load_async_to_lds_b128(
            GPTR(g4 + tid + i * 256), LPTR(s4 + tid + i * 256), 0, 0);
      const float4* gn = (const float4*)Qn + c * (CHUNK_Q / 4);
      float4* sn = (float4*)sQn;
#pragma unroll
      for (int i = 0; i < (CHUNK_Q / 4) / 256; ++i)
        __builtin_amdgcn_global_load_async_to_lds_b128(
            GPTR(gn + tid + i * 256), LPTR(sn + tid + i * 256), 0, 0);
      __builtin_amdgcn_s_wait_asynccnt(0);
    }
#else
    {
      const float4* g4 = (const float4*)Qs + c * CHUNK_Q;
      float4* s4 = (float4*)sQs;
#pragma unroll
      for (int i = 0; i < CHUNK_Q / 256; ++i)
        s4[tid + i * 256] = g4[tid + i * 256];
      const float4* gn = (const float4*)Qn + c * (CHUNK_Q / 4);
      float4* sn = (float4*)sQn;
#pragma unroll
      for (int i = 0; i < (CHUNK_Q / 4) / 256; ++i)
        sn[tid + i * 256] = gn[tid + i * 256];
    }
#endif
    __syncthreads();

    // Byte pointers into LDS; tile stride: A += 256 B, C += 64 B.
    const char* aPtr = (const char*)sQs + (lm << 4) + (half << 3);
    const char* cPtr = (const char*)sQn + (half << 5);
    int ib = c * CHUNK_Q + (half << 3);

    for (int t = 0; t < NTILE; t += 2) {
      v8f D0, D1;
#if USE_WMMA
      v2f A0 = *(const v2f*)(aPtr);
      v8f C0 = *(const v8f*)(cPtr);
      v2f A1 = *(const v2f*)(aPtr + 256);
      v8f C1 = *(const v8f*)(cPtr + 64);
      D0 = __builtin_amdgcn_wmma_f32_16x16x4_f32(false, A0, false, Bop,
                                                 (short)0, C0, false, false);
      D1 = __builtin_amdgcn_wmma_f32_16x16x4_f32(false, A1, false, Bop,
                                                 (short)0, C1, false, false);
#else
#pragma unroll
      for (int r = 0; r < 8; ++r) {
        int row0 = t * 16 + (half << 3) + r;
        const float* q0 = sQs + (row0 << 2);
        D0[r] = sQn[row0] + q0[0]*tb + q0[1]*tx + q0[2]*ty + q0[3]*tz;
        int row1 = row0 + 16;
        const float* q1 = sQs + (row1 << 2);
        D1[r] = sQn[row1] + q1[0]*tb + q1[1]*tx + q1[2]*ty + q1[3]*tz;
      }
#endif
      aPtr += 512; cPtr += 128;

      // quick-reject: only run the insertion network if some lane improves
      float mA = min8(D0);
      if (__any(mA < d2)) {
#pragma unroll
        for (int r = 0; r < 8; ++r)
          insert3(D0[r], ib + r, d0, d1, d2, i0, i1, i2);
      }
      float mB = min8(D1);
      if (__any(mB < d2)) {
#pragma unroll
        for (int r = 0; r < 8; ++r)
          insert3(D1[r], ib + 16 + r, d0, d1, d2, i0, i1, i2);
      }
      ib += 32;
    }
  }

  // Merge the two half-wave candidate sets for each target column.
  {
    float e0 = __shfl_xor(d0, 16); int e0i = __shfl_xor(i0, 16);
    float e1 = __shfl_xor(d1, 16); int e1i = __shfl_xor(i1, 16);
    float e2 = __shfl_xor(d2, 16); int e2i = __shfl_xor(i2, 16);
    insert3(e0, e0i, d0, d1, d2, i0, i1, i2);
    insert3(e1, e1i, d0, d1, d2, i0, i1, i2);
    insert3(e2, e2i, d0, d1, d2, i0, i1, i2);
    // canonicalize: both halves take the low-half merged result
    d0 = __shfl(d0, lm); i0 = __shfl(i0, lm);
    d1 = __shfl(d1, lm); i1 = __shfl(i1, lm);
    d2 = __shfl(d2, lm); i2 = __shfl(i2, lm);
  }

  // True squared distance = score + |t|^2 (clamp tiny negative cancellation).
  float dd0 = fmaxf(d0 + tnorm, 0.0f);
  float dd1 = fmaxf(d1 + tnorm, 0.0f);
  float dd2 = fmaxf(d2 + tnorm, 0.0f);
  float r0 = 1.0f / (dd0 + 1e-8f);
  float r1 = 1.0f / (dd1 + 1e-8f);
  float r2 = 1.0f / (dd2 + 1e-8f);
  float rs = r0 + r1 + r2;
  float w0 = r0 / rs, w1 = r1 / rs, w2 = r2 / rs;

  // three_interpolate: lane half handles 32 of the 64 channels.
  const float4* f0 = (const float4*)(feats + (size_t)i0 * CHANS + half * 32);
  const float4* f1 = (const float4*)(feats + (size_t)i1 * CHANS + half * 32);
  const float4* f2 = (const float4*)(feats + (size_t)i2 * CHANS + half * 32);
  float4* o = (float4*)(out + (size_t)n * CHANS + half * 32);
#pragma unroll
  for (int i = 0; i < 8; ++i) {
    float4 a = f0[i], b = f1[i], cc = f2[i];
    float4 r;
    r.x = w0 * a.x + w1 * b.x + w2 * cc.x;
    r.y = w0 * a.y + w1 * b.y + w2 * cc.y;
    r.z = w0 * a.z + w1 * b.z + w2 * cc.z;
    r.w = w0 * a.w + w1 * b.w + w2 * cc.w;
    o[i] = r;
  }
}

extern "C" void kernel_launch(void* const* d_in, const int* in_sizes, int n_in,
                              void* d_out, int out_size, void* d_ws, size_t ws_size,
                              hipStream_t stream) {
  (void)in_sizes; (void)n_in; (void)out_size; (void)ws_size;
  const float* feats = (const float*)d_in[0];
  const int*   ind   = (const int*)d_in[1];
  const float* pc    = (const float*)d_in[2];
  const int*   bids  = (const int*)d_in[3];

  float* Qs = (float*)d_ws;                 // M_Q*4 floats = 256 KB
  float* Qn = Qs + (size_t)M_Q * 4;         // M_Q floats   =  64 KB
  float* o  = (float*)d_out;

  prep_queries_kernel<<<(M_Q + 255) / 256, 256, 0, stream>>>(ind, Qs, Qn);
  three_nn_interp_kernel<<<N_T / 128, 256, 0, stream>>>(feats, pc, bids, Qs, Qn, o);
}